// Critic_14534169329935
// MI455X (gfx1250) — compile-verified
//
#include <hip/hip_runtime.h>
#include <hip/hip_bf16.h>
#include <math.h>

typedef __attribute__((ext_vector_type(2))) float v2f;
typedef __attribute__((ext_vector_type(8))) float v8f;

#define HID 256
#define G3  768   // 3*HID
#define ROWS 32   // batch rows per workgroup (2 M-tiles share each B fragment)

__device__ __forceinline__ float sigmoidf_(float x) {
    return 1.0f / (1.0f + __expf(-x));
}

// ---------------------------------------------------------------------------
// Pack W [G3, HID] row-major into B-fragment pair order:
//   pack[((k>>1)*G3 + j)*2 + (k&1)] = W[j*HID + k]
// A lane's WMMA B fragment (rows k,k+1 of column j) becomes one aligned b64.
// ---------------------------------------------------------------------------
__global__ void pack_pairs_kernel(const float* __restrict__ in,   // [G3, HID]
                                  float* __restrict__ out) {      // [HID/2, G3, 2]
    int idx = blockIdx.x * blockDim.x + threadIdx.x;
    if (idx < G3 * HID) {
        int j = idx / HID;
        int k = idx - j * HID;
        out[((size_t)(k >> 1) * G3 + j) * 2 + (k & 1)] = in[idx];
    }
}

// ---------------------------------------------------------------------------
// GRU over N steps for 32 batch rows per workgroup.
// x_t = [label_t, data_t] (2-d input). gh = h @ Whh^T via V_WMMA_F32_16X16X4_F32.
// h kept resident in LDS across all time steps. Each wave computes two 16x16
// M-tiles per B fragment (one b64 B load feeds two WMMAs -> 2x B reuse).
// ---------------------------------------------------------------------------
__global__ __launch_bounds__(256) void gru_seq_kernel(
    const float* __restrict__ data,   // [B, N, D]
    const float* __restrict__ label,  // [B, N, D]
    const float* __restrict__ Wih,    // [3H, 2]
    const float* __restrict__ Bpack,  // packed Whh^T pairs
    const float* __restrict__ bih,    // [3H]
    const float* __restrict__ bhh,    // [3H]
    float* __restrict__ outH,         // [B*D, H]
    int N, int D) {
    __shared__ float hS[ROWS * HID];   // 32 KB
    __shared__ float gS[ROWS * G3];    // 96 KB
    __shared__ float xS[ROWS * 2];

    const int tid  = threadIdx.x;
    const int r0   = blockIdx.x * ROWS;
    const int lane = tid & 31;
    const int wave = tid >> 5;
    const int col  = lane & 15;            // N-column within tile / A row
    const int koff = (lane >> 4) << 1;     // lanes 16-31 carry K+2,K+3
    const int mbase = (lane >> 4) << 3;    // C/D rows: M = v + 8*(lane>=16)

    // h0 = 0
    for (int i = tid; i < ROWS * HID; i += 256) hS[i] = 0.0f;

    // time-invariant gate weights for this thread's hidden index k = tid
    const int k = tid;
    const float w_r0 = Wih[k * 2 + 0],           w_r1 = Wih[k * 2 + 1];
    const float w_z0 = Wih[(k + HID) * 2 + 0],   w_z1 = Wih[(k + HID) * 2 + 1];
    const float w_n0 = Wih[(k + 2*HID) * 2 + 0], w_n1 = Wih[(k + 2*HID) * 2 + 1];
    const float b_ir = bih[k], b_iz = bih[k + HID], b_in = bih[k + 2*HID];
    const float b_hr = bhh[k], b_hz = bhh[k + HID], b_hn = bhh[k + 2*HID];

    for (int t = 0; t < N; ++t) {
        if (tid < ROWS) {
            int m  = r0 + tid;
            int b  = m / D;
            int dd = m - b * D;
            size_t idx = ((size_t)b * N + t) * D + dd;
            xS[tid * 2 + 0] = label[idx];   // x[...,0] = label
            xS[tid * 2 + 1] = data[idx];    // x[...,1] = data
        }
        __syncthreads();

        // gh = hS(32xH) @ Whh^T(Hx3H): each wave owns 6 N-tiles x 2 M-tiles
        for (int i = 0; i < 6; ++i) {
            const int j0 = ((wave * 6 + i) << 4);
            // prefetch next N-tile's B pairs into the cache hierarchy
            if (i < 5)
                __builtin_prefetch(Bpack + ((size_t)(koff >> 1) * G3 + j0 + 16 + col) * 2, 0, 1);
            v8f acc0 = {};
            v8f acc1 = {};
            #pragma unroll 8
            for (int k0 = 0; k0 < HID; k0 += 4) {
                v2f a0, a1;
                v2f bf = *(const v2f*)(Bpack +
                        ((size_t)((k0 + koff) >> 1) * G3 + j0 + col) * 2);
                a0.x = hS[col * HID + k0 + koff];
                a0.y = hS[col * HID + k0 + koff + 1];
                a1.x = hS[(16 + col) * HID + k0 + koff];
                a1.y = hS[(16 + col) * HID + k0 + koff + 1];
                acc0 = __builtin_amdgcn_wmma_f32_16x16x4_f32(
                        false, a0, false, bf, (short)0, acc0, false, false);
                acc1 = __builtin_amdgcn_wmma_f32_16x16x4_f32(
                        false, a1, false, bf, (short)0, acc1, false, false);
            }
            #pragma unroll
            for (int v = 0; v < 8; ++v) {
                gS[(mbase + v) * G3 + j0 + col]        = acc0[v];
                gS[(16 + mbase + v) * G3 + j0 + col]   = acc1[v];
            }
        }
        __syncthreads();

        // gate math: thread k handles column k of all 32 rows (in-place h update)
        for (int m = 0; m < ROWS; ++m) {
            float x0 = xS[m * 2 + 0], x1 = xS[m * 2 + 1];
            float ir  = w_r0 * x0 + w_r1 * x1 + b_ir;
            float iz  = w_z0 * x0 + w_z1 * x1 + b_iz;
            float inn = w_n0 * x0 + w_n1 * x1 + b_in;
            float hr = gS[m * G3 + k]           + b_hr;
            float hz = gS[m * G3 + k + HID]     + b_hz;
            float hn = gS[m * G3 + k + 2*HID]   + b_hn;
            float r = sigmoidf_(ir + hr);
            float z = sigmoidf_(iz + hz);
            float n = tanhf(inn + r * hn);
            float hp = hS[m * HID + k];
            hS[m * HID + k] = (1.0f - z) * n + z * hp;
        }
        __syncthreads();
    }

    for (int m = 0; m < ROWS; ++m)
        outH[(size_t)(r0 + m) * HID + tid] = hS[m * HID + tid];
}

// ---------------------------------------------------------------------------
// Critic GRU single step with h0 = 0  =>  gh = bhh exactly.
// gi = dec @ Wih_g^T via WMMA, then rnn = (1-z)*n.
// ---------------------------------------------------------------------------
#define CROWS 16
__global__ __launch_bounds__(256) void critic_step_kernel(
    const float* __restrict__ dec,    // [B, H]
    const float* __restrict__ Bpack,  // packed Wih_g^T pairs
    const float* __restrict__ bih,    // [3H]
    const float* __restrict__ bhh,    // [3H]
    float* __restrict__ rnn) {        // [B, H]
    __shared__ float aS[CROWS * HID];
    __shared__ float gS[CROWS * G3];

    const int tid  = threadIdx.x;
    const int r0   = blockIdx.x * CROWS;
    const int lane = tid & 31;
    const int wave = tid >> 5;
    const int col  = lane & 15;
    const int koff = (lane >> 4) << 1;
    const int mbase = (lane >> 4) << 3;

    for (int i = tid; i < CROWS * HID; i += 256)
        aS[i] = dec[(size_t)r0 * HID + i];
    __syncthreads();

    for (int i = 0; i < 6; ++i) {
        const int j0 = ((wave * 6 + i) << 4);
        v8f acc = {};
        #pragma unroll 8
        for (int k0 = 0; k0 < HID; k0 += 4) {
            v2f a;
            v2f bf = *(const v2f*)(Bpack +
                    ((size_t)((k0 + koff) >> 1) * G3 + j0 + col) * 2);
            a.x  = aS[col * HID + k0 + koff];
            a.y  = aS[col * HID + k0 + koff + 1];
            acc = __builtin_amdgcn_wmma_f32_16x16x4_f32(
                    false, a, false, bf, (short)0, acc, false, false);
        }
        #pragma unroll
        for (int v = 0; v < 8; ++v)
            gS[(mbase + v) * G3 + j0 + col] = acc[v];
    }
    __syncthreads();

    const int k = tid;
    const float b_ir = bih[k], b_iz = bih[k + HID], b_in = bih[k + 2*HID];
    const float b_hr = bhh[k], b_hz = bhh[k + HID], b_hn = bhh[k + 2*HID];
    for (int m = 0; m < CROWS; ++m) {
        float r = sigmoidf_(gS[m * G3 + k]         + b_ir + b_hr);
        float z = sigmoidf_(gS[m * G3 + k + HID]   + b_iz + b_hz);
        float n = tanhf(gS[m * G3 + k + 2*HID] + b_in + r * b_hn);
        rnn[(size_t)(r0 + m) * HID + k] = (1.0f - z) * n;   // + z*0
    }
}

// ---------------------------------------------------------------------------
// 3 attention process blocks + FC head. One workgroup per batch element b:
// everything (enc slice, rnn_out) is b-local, so the whole chain runs in LDS.
// ---------------------------------------------------------------------------
__global__ __launch_bounds__(256) void attn_fc_kernel(
    const float* __restrict__ enc,    // [B, D=8, H]  (== encoder_hidden[b,h,d] at [b,d,h])
    const float* __restrict__ rnn_in, // [B, H]
    const float* __restrict__ att_v,  // [H]
    const float* __restrict__ att_W,  // [H, 2H]
    const float* __restrict__ fc_W1,  // [H, H]
    const float* __restrict__ fc_b1,  // [H]
    const float* __restrict__ fc_W2,  // [1, H]
    const float* __restrict__ fc_b2,  // [1]
    float* __restrict__ out) {        // [B]
    __shared__ float encL[8 * HID];
    __shared__ float rnnL[HID];
    __shared__ float sbuf[8 * HID];
    __shared__ float scores[8];
    __shared__ float prob[8];
    __shared__ float red[HID];

    const int b = blockIdx.x;
    const int t = threadIdx.x;

    for (int i = t; i < 8 * HID; i += 256)
        encL[i] = enc[(size_t)b * (8 * HID) + i];
    rnnL[t] = rnn_in[(size_t)b * HID + t];
    __syncthreads();

    const float av = att_v[t];
    const float* wrow = att_W + (size_t)t * (2 * HID);

    for (int blk = 0; blk < 3; ++blk) {
        // s2 = att_W[t, H:2H] . rnn   (d-independent half of hcat)
        float s2 = 0.0f;
        for (int h = 0; h < HID; ++h) s2 += wrow[HID + h] * rnnL[h];
        // u[t,d] and score contribution
        for (int d = 0; d < 8; ++d) {
            float acc = s2;
            const float* e = encL + d * HID;
            for (int h = 0; h < HID; ++h) acc += wrow[h] * e[h];
            sbuf[d * HID + t] = av * tanhf(acc);
        }
        __syncthreads();
        if (t < 8) {
            float s = 0.0f;
            for (int i = 0; i < HID; ++i) s += sbuf[t * HID + i];
            scores[t] = s;
        }
        __syncthreads();
        if (t == 0) {
            float mx = scores[0];
            for (int d = 1; d < 8; ++d) mx = fmaxf(mx, scores[d]);
            float sum = 0.0f;
            for (int d = 0; d < 8; ++d) { float e = __expf(scores[d] - mx); prob[d] = e; sum += e; }
            float inv = 1.0f / sum;
            for (int d = 0; d < 8; ++d) prob[d] *= inv;
        }
        __syncthreads();
        float rn = 0.0f;
        for (int d = 0; d < 8; ++d) rn += prob[d] * encL[d * HID + t];
        rnnL[t] = rn;
        __syncthreads();
    }

    // FC head: h1 = relu(rnn @ W1^T + b1); out = W2 . h1 + b2
    float acc = fc_b1[t];
    const float* w1 = fc_W1 + (size_t)t * HID;
    for (int h = 0; h < HID; ++h) acc += w1[h] * rnnL[h];
    float h1 = acc > 0.0f ? acc : 0.0f;
    red[t] = fc_W2[t] * h1;
    __syncthreads();
    for (int s = 128; s > 0; s >>= 1) {
        if (t < s) red[t] += red[t + s];
        __syncthreads();
    }
    if (t == 0) out[b] = red[0] + fc_b2[0];
}

// ---------------------------------------------------------------------------
extern "C" void kernel_launch(void* const* d_in, const int* in_sizes, int n_in,
                              void* d_out, int out_size, void* d_ws, size_t ws_size,
                              hipStream_t stream) {
    const int B = 256, N = 128, D = 8;

    const float* enc_data = (const float*)d_in[0];
    const float* enc_lab  = (const float*)d_in[1];
    const float* dec_data = (const float*)d_in[2];
    const float* dec_lab  = (const float*)d_in[3];
    const float* Wih_e = (const float*)d_in[4];
    const float* Whh_e = (const float*)d_in[5];
    const float* bih_e = (const float*)d_in[6];
    const float* bhh_e = (const float*)d_in[7];
    const float* Wih_d = (const float*)d_in[8];
    const float* Whh_d = (const float*)d_in[9];
    const float* bih_d = (const float*)d_in[10];
    const float* bhh_d = (const float*)d_in[11];
    const float* Wih_g = (const float*)d_in[12];
    // d_in[13] (Whh_g) is mathematically unused: h0 == 0 for the critic GRU.
    const float* bih_g = (const float*)d_in[14];
    const float* bhh_g = (const float*)d_in[15];
    const float* att_v = (const float*)d_in[16];
    const float* att_W = (const float*)d_in[17];
    const float* fc_W1 = (const float*)d_in[18];
    const float* fc_b1 = (const float*)d_in[19];
    const float* fc_W2 = (const float*)d_in[20];
    const float* fc_b2 = (const float*)d_in[21];

    // workspace layout (floats)
    float* ws = (float*)d_ws;
    const size_t TSZ = (size_t)HID * G3;            // 196608
    float* Bpack_e = ws;                            // packed Whh_e^T pairs
    float* Bpack_d = ws + TSZ;                      // packed Whh_d^T pairs
    float* Bpack_g = ws + 2 * TSZ;                  // packed Wih_g^T pairs
    float* encH    = ws + 3 * TSZ;                  // [B*D, H]
    float* decH    = encH + (size_t)B * D * HID;    // [B, H]
    float* rnnH    = decH + (size_t)B * HID;        // [B, H]

    const int RC = G3 * HID;
    pack_pairs_kernel<<<(RC + 255) / 256, 256, 0, stream>>>(Whh_e, Bpack_e);
    pack_pairs_kernel<<<(RC + 255) / 256, 256, 0, stream>>>(Whh_d, Bpack_d);
    pack_pairs_kernel<<<(RC + 255) / 256, 256, 0, stream>>>(Wih_g, Bpack_g);

    // encoder: B*D = 2048 rows -> 64 workgroups
    gru_seq_kernel<<<(B * D) / ROWS, 256, 0, stream>>>(
        enc_data, enc_lab, Wih_e, Bpack_e, bih_e, bhh_e, encH, N, D);
    // decoder: B rows (D=1) -> 8 workgroups
    gru_seq_kernel<<<B / ROWS, 256, 0, stream>>>(
        dec_data, dec_lab, Wih_d, Bpack_d, bih_d, bhh_d, decH, N, 1);
    // critic single step
    critic_step_kernel<<<B / CROWS, 256, 0, stream>>>(decH, Bpack_g, bih_g, bhh_g, rnnH);
    // attention x3 + FC
    attn_fc_kernel<<<B, 256, 0, stream>>>(encH, rnnH, att_v, att_W,
                                          fc_W1, fc_b1, fc_W2, fc_b2, (float*)d_out);
}